// PerJointLoss_43026982371528
// MI455X (gfx1250) — compile-verified
//
#include <hip/hip_runtime.h>
#include <math.h>

// Problem geometry (compile-time constants from the reference).
#define NJ    24
#define JD    3
#define TS    120
#define NB    16384
#define COLS  (NJ * JD * TS)   // 8640 floats per batch, contiguous
#define VCOLS (COLS / 4)       // 2160 float4 columns per batch
#define NCHUNK 32              // batch chunks (deterministic partials)
#define CHUNK  (NB / NCHUNK)   // 512 batches per chunk
#define NOUT   (NJ * TS)       // 2880 (j,t) norms

typedef __attribute__((ext_vector_type(2))) float v2f;
typedef __attribute__((ext_vector_type(8))) float v8f;

// ---------------------------------------------------------------------------
// Kernel 1: streaming squared-diff accumulation. Memory-bound (1.13 GB read);
// each thread owns 4 consecutive columns (one float4) and strides over 512
// batches. Consecutive lanes -> consecutive 16B -> fully coalesced b128 loads.
// Writes per-chunk partial sums part[chunk][8640] (no atomics -> deterministic).
// ---------------------------------------------------------------------------
__global__ void __launch_bounds__(256) pjl_partial(
    const float* __restrict__ src, const float* __restrict__ tgt,
    float* __restrict__ part) {
  const int vcol = blockIdx.x * 256 + threadIdx.x;
  if (vcol >= VCOLS) return;

  const float4* __restrict__ s4 = (const float4*)src;
  const float4* __restrict__ t4 = (const float4*)tgt;
  long base = (long)blockIdx.y * CHUNK * VCOLS + vcol;

  float a0 = 0.f, a1 = 0.f, a2 = 0.f, a3 = 0.f;
#pragma unroll 4
  for (int i = 0; i < CHUNK; ++i) {
    float4 sv = s4[base];
    float4 tv = t4[base];
    float d0 = sv.x - tv.x;
    float d1 = sv.y - tv.y;
    float d2 = sv.z - tv.z;
    float d3 = sv.w - tv.w;
    a0 = fmaf(d0, d0, a0);
    a1 = fmaf(d1, d1, a1);
    a2 = fmaf(d2, d2, a2);
    a3 = fmaf(d3, d3, a3);
    base += VCOLS;
  }
  float4 r;
  r.x = a0; r.y = a1; r.z = a2; r.w = a3;
  ((float4*)part)[(size_t)blockIdx.y * VCOLS + vcol] = r;
}

// ---------------------------------------------------------------------------
// Kernel 2: fold 32 chunk-partials and the 3 joint-dims per (j,t), sqrt.
// 2880 outputs; loads are coalesced along t.
// ---------------------------------------------------------------------------
__global__ void __launch_bounds__(256) pjl_fold(
    const float* __restrict__ part, float* __restrict__ vals) {
  const int i = blockIdx.x * 256 + threadIdx.x;
  if (i >= NOUT) return;
  const int j = i / TS;
  const int t = i - j * TS;
  const int col = j * (JD * TS) + t;   // row d=0 of joint j at timestep t
  float s = 0.f;
#pragma unroll 4
  for (int ch = 0; ch < NCHUNK; ++ch) {
    const float* __restrict__ p = part + (size_t)ch * COLS + col;
    s += p[0] + p[TS] + p[2 * TS];
  }
  vals[i] = sqrtf(s);
}

// ---------------------------------------------------------------------------
// Kernel 3: sum 2880 f32 values on the matrix unit (wave32, EXEC all-1s).
// V_WMMA_F32_16X16X4_F32 with B = ones: D[m,n] = sum_k A[m,k] + C[m,n].
// Each iteration feeds 64 values through A (2 VGPRs/lane); 45 iterations
// cover 2880 exactly. C[m,n] then holds row-m's running sum (replicated over
// n). Per the ISA C/D layout, lane L<16 holds rows 0..7, lane L>=16 rows
// 8..15, so: per-lane sum of the 8 C components + shfl_xor(16) = total.
// Pure f32 the whole way -> exact, no precision compromise.
// ---------------------------------------------------------------------------
__global__ void __launch_bounds__(32) pjl_wmma_reduce(
    const float* __restrict__ vals, float* __restrict__ out) {
  const int lane = threadIdx.x;   // 32 threads, no divergence: EXEC all ones
  v8f c = {};
  v2f ones;
  ones.x = 1.0f;
  ones.y = 1.0f;

  for (int base = 0; base < NOUT; base += 64) {
    v2f a;
    a.x = vals[base + lane];
    a.y = vals[base + 32 + lane];
    // (neg_a, A, neg_b, B, c_mod, C, reuse_a, reuse_b)
    c = __builtin_amdgcn_wmma_f32_16x16x4_f32(
        false, a, false, ones, (short)0, c, false, false);
  }

  float s = c[0] + c[1] + c[2] + c[3] + c[4] + c[5] + c[6] + c[7];
  s += __shfl_xor(s, 16, 32);     // combine rows 0-7 half with rows 8-15 half
  if (lane == 0) out[0] = s;
}

// ---------------------------------------------------------------------------
// Launch: partial -> fold -> wmma-reduce, all on `stream` (in-order).
// d_ws layout: [ part: 32*8640 f32 | vals: 2880 f32 ]  (~1.07 MB)
// Every d_ws/d_out byte consumed is produced first each call.
// ---------------------------------------------------------------------------
extern "C" void kernel_launch(void* const* d_in, const int* in_sizes, int n_in,
                              void* d_out, int out_size, void* d_ws, size_t ws_size,
                              hipStream_t stream) {
  const float* src = (const float*)d_in[0];
  const float* tgt = (const float*)d_in[1];
  float* part = (float*)d_ws;
  float* vals = part + (size_t)NCHUNK * COLS;
  float* out  = (float*)d_out;

  dim3 g1((VCOLS + 255) / 256, NCHUNK);     // 9 x 32 blocks
  pjl_partial<<<g1, 256, 0, stream>>>(src, tgt, part);

  pjl_fold<<<(NOUT + 255) / 256, 256, 0, stream>>>(part, vals);

  pjl_wmma_reduce<<<1, 32, 0, stream>>>(vals, out);
}